// RNN_mLor_2D_46797963657658
// MI455X (gfx1250) — compile-verified
//
#include <hip/hip_runtime.h>
#include <math.h>
#include <stdint.h>

// Grid geometry (fixed by the reference)
#define NEg   421            // TX+1 == TY+1
#define TXg   420
#define TYg   420
#define N1g   (421*421)      // Ez/Eold/Jz/Jold element count
#define NHXg  (421*420)      // Hx
#define NHYg  (420*421)      // Hy
#define SRCI  210
#define SRCJ  210
#define TILE  32
#define NT    14             // ceil(421/32)

#if __has_builtin(__builtin_amdgcn_tensor_load_to_lds)
#define USE_TDM 1
#else
#define USE_TDM 0
#endif

typedef uint32_t v4u __attribute__((ext_vector_type(4)));
typedef int      v8i __attribute__((ext_vector_type(8)));
typedef int      v4i __attribute__((ext_vector_type(4)));

// Low 32 bits of a generic pointer into LDS == byte offset from LDS base.
__device__ __forceinline__ uint32_t lds_base_off(const void* p) {
  return (uint32_t)(uintptr_t)p;
}

// ---------------- CDNA5 Tensor Data Mover: one-instruction 2-D tile load ----------------
// D# layout per CDNA5 ISA ch.8: group0 = {count/type, lds_addr, global_addr},
// group1 = {data_size, tensor_dim0/1 (remaining extent => HW zero-fills OOB),
//           tile_dim0/1, tensor_dim0_stride}.  2-D only: further groups zero.
// clang-23 builtin arity: (v4u g0, v8i g1, v4i g2, v4i g3, v8i g4, i32 cpol).
__device__ __forceinline__ void tdm_load_2d(
    uint32_t lds_byte, const void* gaddr,
    uint32_t dim0_rem, uint32_t dim1_rem,   // tensor extent remaining from tile start (elems)
    uint32_t tile_w, uint32_t tile_h,       // tile dims (elems)
    uint32_t row_stride)                    // elems between rows
{
#if USE_TDM
  uint64_t ga = (uint64_t)(uintptr_t)gaddr;
  v4u g0;
  g0.x = 1u;                                                 // count=1 (valid user D#)
  g0.y = lds_byte;                                           // bits 63:32  lds_addr
  g0.z = (uint32_t)(ga & 0xFFFFFFFFu);                       // bits 95:64  global_addr lo
  g0.w = (uint32_t)((ga >> 32) & 0x1FFFFFFu) | (2u << 30);   // addr hi | type=2
  v8i g1;
  g1[0] = (int)(2u << 16);                                   // data_size = 4 bytes
  g1[1] = (int)((dim0_rem & 0xFFFFu) << 16);                 // tensor_dim0 lo16 @ bits 63:48
  g1[2] = (int)(((dim0_rem >> 16) & 0xFFFFu) | ((dim1_rem & 0xFFFFu) << 16));
  g1[3] = (int)(((dim1_rem >> 16) & 0xFFFFu) | (tile_w << 16)); // tile_dim0 @ 127:112
  g1[4] = (int)(tile_h & 0xFFFFu);                           // tile_dim1; tile_dim2=0 (2-D)
  g1[5] = (int)row_stride;                                   // tensor_dim0_stride lo32
  g1[6] = 0;                                                 // stride hi | dim1_stride lo
  g1[7] = 0;
  v4i gz4 = {0, 0, 0, 0};
  v8i gz8 = {0, 0, 0, 0, 0, 0, 0, 0};
  __builtin_amdgcn_tensor_load_to_lds(g0, g1, gz4, gz4, gz8, 0);
#endif
  (void)lds_byte; (void)gaddr; (void)dim0_rem; (void)dim1_rem;
  (void)tile_w; (void)tile_h; (void)row_stride;
}

__device__ __forceinline__ void wait_tensor0() {
#if __has_builtin(__builtin_amdgcn_s_wait_tensorcnt)
  __builtin_amdgcn_s_wait_tensorcnt(0);
#else
  asm volatile("s_wait_tensorcnt 0" ::: "memory");
#endif
}

// ---------------- fallback: per-lane async global->LDS (proven to assemble) ----------------
__device__ __forceinline__ void async_g2l_b32(uint32_t lds_off, const float* g) {
  uint64_t ga = (uint64_t)(uintptr_t)g;
  asm volatile("global_load_async_to_lds_b32 %0, %1, off"
               :: "v"(lds_off), "v"(ga)
               : "memory");
}
__device__ __forceinline__ void wait_async0() {
#if __has_builtin(__builtin_amdgcn_s_wait_asynccnt)
  __builtin_amdgcn_s_wait_asynccnt(0);
#else
  asm volatile("s_wait_asynccnt 0" ::: "memory");
#endif
}

// ---------------- init: zero state + build separable damping tables ----------------
__global__ __launch_bounds__(256) void init_kernel(
    float* __restrict__ ezA, float* __restrict__ ezB,
    float* __restrict__ jP,  float* __restrict__ jQ,
    float* __restrict__ Hx,  float* __restrict__ Hy,
    const float* __restrict__ sig_ex, const float* __restrict__ sig_ey,
    const float* __restrict__ sig_hx, const float* __restrict__ sig_hy,
    float* __restrict__ ex_e, float* __restrict__ ey_e,
    float* __restrict__ ex_h, float* __restrict__ ey_h,
    float* __restrict__ hx_h, float* __restrict__ hy_h,
    float dt_eps, float dt_mu)
{
  int idx = blockIdx.x * blockDim.x + threadIdx.x;
  if (idx < N1g) { ezA[idx] = 0.f; ezB[idx] = 0.f; jP[idx] = 0.f; jQ[idx] = 0.f; }
  if (idx < NHXg) Hx[idx] = 0.f;
  if (idx < NHYg) Hy[idx] = 0.f;
  if (idx < NEg) {
    float sx = sig_ex[idx], sy = sig_ey[idx];
    ex_e[idx] = expf(-sx * dt_eps);
    ex_h[idx] = expf(-sx * dt_mu);
    ey_e[idx] = expf(-sy * dt_eps);
    ey_h[idx] = expf(-sy * dt_mu);
  }
  if (idx < TXg) hx_h[idx] = expf(-sig_hx[idx] * dt_mu);
  if (idx < TYg) hy_h[idx] = expf(-sig_hy[idx] * dt_mu);
}

// ---------------- H half-step: Hx,Hy from Ez (Ez tile staged via TDM) ----------------
__global__ __launch_bounds__(256) void hstep(
    const float* __restrict__ Ez,
    float* __restrict__ Hx, float* __restrict__ Hy,
    const float* __restrict__ dbhx, const float* __restrict__ dbhy,
    const float* __restrict__ ex_h, const float* __restrict__ ey_h,
    const float* __restrict__ hx_h, const float* __restrict__ hy_h)
{
  __shared__ float ez_t[33 * 33];               // Ez[i0 .. i0+32][j0 .. j0+32]
  const int i0 = blockIdx.y * TILE;
  const int j0 = blockIdx.x * TILE;
  const int tx = threadIdx.x, ty = threadIdx.y; // block (32,8)
  const uint32_t lb = lds_base_off(ez_t);

#if USE_TDM
  if (ty == 0) {                                // one wave issues the DMA
    tdm_load_2d(lb, Ez + i0 * NEg + j0,
                (uint32_t)(NEg - j0), (uint32_t)(NEg - i0),   // HW zero-fills OOB
                33u, 33u, (uint32_t)NEg);
    wait_tensor0();
  }
  __syncthreads();
#else
  {
    const int tid = ty * 32 + tx;
    for (int e = tid; e < 33 * 33; e += 256) {
      int r = e / 33, c = e - r * 33;
      int gi = min(i0 + r, TXg);
      int gj = min(j0 + c, TYg);
      async_g2l_b32(lb + 4u * (uint32_t)e, Ez + gi * NEg + gj);
    }
    wait_async0();
    __syncthreads();
  }
#endif

#pragma unroll
  for (int k = 0; k < 4; ++k) {
    int r  = ty + 8 * k;
    int gi = i0 + r;
    int gj = j0 + tx;
    float e00 = ez_t[r * 33 + tx];
    float e01 = ez_t[r * 33 + tx + 1];
    float e10 = ez_t[(r + 1) * 33 + tx];
    if (gi <= TXg && gj < TYg) {               // Hx: (421,420)
      int idx = gi * TYg + gj;
      Hx[idx] = ex_h[gi] * hy_h[gj] * (Hx[idx] - dbhx[idx] * (e01 - e00));
    }
    if (gi < TXg && gj <= TYg) {               // Hy: (420,421)
      int idx = gi * NEg + gj;
      Hy[idx] = hx_h[gi] * ey_h[gj] * (Hy[idx] + dbhy[idx] * (e10 - e00));
    }
  }
}

// ---------------- E half-step: Ez,Jz pointwise; curl(H) tiles staged via TDM ----------------
__global__ __launch_bounds__(256) void estep(
    const float* __restrict__ EzC, float* __restrict__ EzO,   // EzO: Eold in, Ez_new out
    const float* __restrict__ JzC, float* __restrict__ JzO,   // JzO: Jold in, Jz_new out
    const float* __restrict__ Hx, const float* __restrict__ Hy,
    const float* __restrict__ C1, const float* __restrict__ C2,
    const float* __restrict__ Cbdx, const float* __restrict__ Cbdy,
    const float* __restrict__ ex_e, const float* __restrict__ ey_e,
    const float* __restrict__ pCa, const float* __restrict__ pCb,
    const float* __restrict__ pCc, const float* __restrict__ pCd,
    const float* __restrict__ pCe,
    const float* __restrict__ src, int sidx)
{
  __shared__ float hx_t[32 * 33];   // Hx[i0+r][j0-1+c], c in 0..32
  __shared__ float hy_t[33 * 32];   // Hy[i0-1+r][j0+c], r in 0..32
  const int i0 = blockIdx.y * TILE;
  const int j0 = blockIdx.x * TILE;
  const int tx = threadIdx.x, ty = threadIdx.y;
  const uint32_t bx = lds_base_off(hx_t);
  const uint32_t by = lds_base_off(hy_t);

#if USE_TDM
  if (ty == 0) {
    // Hx tile: rows i0..i0+31 (of 421), cols j0-1..j0+31 (of 420), stride 420.
    // Left halo at j0==0 reads 1 stray (in-workspace) element, consumed only by
    // the masked boundary column.  Right/bottom OOB is zero-filled by TDM.
    tdm_load_2d(bx, Hx + i0 * TYg + (j0 - 1),
                (uint32_t)(TYg - (j0 - 1)), (uint32_t)((TXg + 1) - i0),
                33u, 32u, (uint32_t)TYg);
    // Hy tile: rows i0-1..i0+31 (of 420), cols j0..j0+31 (of 421), stride 421.
    tdm_load_2d(by, Hy + (i0 - 1) * NEg + j0,
                (uint32_t)(NEg - j0), (uint32_t)(TXg - (i0 - 1)),
                32u, 33u, (uint32_t)NEg);
    wait_tensor0();
  }
  __syncthreads();
#else
  {
    const int tid = ty * 32 + tx;
    for (int e = tid; e < 32 * 33; e += 256) {
      int r = e / 33, c = e - r * 33;
      int gi = min(i0 + r, TXg);
      int gj = min(max(j0 - 1 + c, 0), TYg - 1);
      async_g2l_b32(bx + 4u * (uint32_t)e, Hx + gi * TYg + gj);
    }
    for (int e = tid; e < 33 * 32; e += 256) {
      int r = e / 32, c = e - r * 32;
      int gi = min(max(i0 - 1 + r, 0), TXg - 1);
      int gj = min(j0 + c, TYg);
      async_g2l_b32(by + 4u * (uint32_t)e, Hy + gi * NEg + gj);
    }
    wait_async0();
    __syncthreads();
  }
#endif

  const float ca = pCa[0], cb = pCb[0], cc = pCc[0], cd = pCd[0], ce = pCe[0];

#pragma unroll
  for (int k = 0; k < 4; ++k) {
    int r  = ty + 8 * k;
    int gi = i0 + r;
    int gj = j0 + tx;
    if (gi > TXg || gj > TYg) continue;
    int idx = gi * NEg + gj;

    float ez = EzC[idx];
    float eo = EzO[idx];
    float jz = JzC[idx];
    float jo = JzO[idx];

    float curlHy = hy_t[(r + 1) * 32 + tx] - hy_t[r * 32 + tx];  // Hy[i][j]-Hy[i-1][j]
    float curlHx = hx_t[r * 33 + tx + 1] - hx_t[r * 33 + tx];    // Hx[i][j]-Hx[i][j-1]

    float phi = (ca + 1.f) * jz + cb * jo + cd * ez + ce * eo;
    float de  = ex_e[gi] * ey_e[gj];

    bool bnd = (gi == 0) | (gi == TXg) | (gj == 0) | (gj == TYg);
    float ezn = bnd ? 0.f
              : de * (C1[idx] * ez + Cbdx[idx] * curlHy - Cbdy[idx] * curlHx - C2[idx] * phi);
    if (gi == SRCI && gj == SRCJ) ezn += src[sidx];

    float jzn = ca * jz + cb * jo + cc * ezn + cd * ez + ce * eo;

    EzO[idx] = ezn;
    JzO[idx] = jzn;
  }
}

// ---------------- host-side orchestration ----------------
extern "C" void kernel_launch(void* const* d_in, const int* in_sizes, int n_in,
                              void* d_out, int out_size, void* d_ws, size_t ws_size,
                              hipStream_t stream) {
  const float* src    = (const float*)d_in[0];
  const float* C1     = (const float*)d_in[1];
  const float* C2     = (const float*)d_in[2];
  const float* Cbdx   = (const float*)d_in[3];
  const float* Cbdy   = (const float*)d_in[4];
  const float* dbhx   = (const float*)d_in[5];
  const float* dbhy   = (const float*)d_in[6];
  const float* Ca     = (const float*)d_in[7];
  const float* Cb     = (const float*)d_in[8];
  const float* Cc     = (const float*)d_in[9];
  const float* Cd     = (const float*)d_in[10];
  const float* Ce     = (const float*)d_in[11];
  const float* sig_ex = (const float*)d_in[12];
  const float* sig_ey = (const float*)d_in[13];
  const float* sig_hx = (const float*)d_in[14];
  const float* sig_hy = (const float*)d_in[15];
  const int nsteps = in_sizes[0];               // scan length == len(src)

  // workspace layout (Hx intentionally NOT first: halo reads at -1 stay in-bounds)
  float* w   = (float*)d_ws;
  float* ezA = w; w += N1g;
  float* ezB = w; w += N1g;
  float* jP  = w; w += N1g;
  float* jQ  = w; w += N1g;
  float* Hx  = w; w += NHXg;
  float* Hy  = w; w += NHYg;
  float* ex_e = w; w += NEg;
  float* ey_e = w; w += NEg;
  float* ex_h = w; w += NEg;
  float* ey_h = w; w += NEg;
  float* hx_h = w; w += TXg;
  float* hy_h = w; w += TYg;

  // physical constants (match the reference's double-precision derivation)
  const double PI   = 3.14159265358979323846;
  const double EPS0 = 1e-9 / 36.0 / PI;
  const double MU0  = 4.0 * PI * 1e-7;
  const double C0   = 1.0 / sqrt(MU0 * EPS0);
  const double DX   = 2.5e-8, DY = 2.5e-8;
  const double DT   = 0.99 / C0 / sqrt(1.0 / (DX * DX) + 1.0 / (DY * DY));
  const float dt_eps = (float)(DT / EPS0);
  const float dt_mu  = (float)(DT / MU0);

  init_kernel<<<(N1g + 255) / 256, 256, 0, stream>>>(
      ezA, ezB, jP, jQ, Hx, Hy, sig_ex, sig_ey, sig_hx, sig_hy,
      ex_e, ey_e, ex_h, ey_h, hx_h, hy_h, dt_eps, dt_mu);

  dim3 grid(NT, NT), block(32, 8);
  float* ezC = ezA; float* ezO = ezB;
  float* jC  = jP;  float* jO  = jQ;
  for (int s = 0; s < nsteps; ++s) {
    hstep<<<grid, block, 0, stream>>>(ezC, Hx, Hy, dbhx, dbhy, ex_h, ey_h, hx_h, hy_h);
    estep<<<grid, block, 0, stream>>>(ezC, ezO, jC, jO, Hx, Hy, C1, C2, Cbdx, Cbdy,
                                      ex_e, ey_e, Ca, Cb, Cc, Cd, Ce, src, s);
    float* t = ezC; ezC = ezO; ezO = t;
    t = jC; jC = jO; jO = t;
  }

  (void)hipMemcpyAsync(d_out, ezC, (size_t)N1g * sizeof(float),
                       hipMemcpyDeviceToDevice, stream);
}